// ImpulseSolverLCP_74096775790651
// MI455X (gfx1250) — compile-verified
//
#include <hip/hip_runtime.h>

#define KREG 1e-3f
#define SS 132   // padded LDS row stride for the 128x128 Schur matrix

typedef float v2f __attribute__((ext_vector_type(2)));
typedef float v8f __attribute__((ext_vector_type(8)));
typedef unsigned int u32x4 __attribute__((ext_vector_type(4)));
typedef int i32x8 __attribute__((ext_vector_type(8)));
typedef int i32x4 __attribute__((ext_vector_type(4)));

// ---------------------------------------------------------------------------
// Kernel 1 (one workgroup): M = inv(Minv); W = inv(M + reg*I); P = W*M.
// Gauss-Jordan on 64x64 (Minv and M+reg*I are SPD -> no pivoting needed).
// ---------------------------------------------------------------------------
__global__ __launch_bounds__(256) void setup_kernel(const float* __restrict__ Minv,
                                                    float* __restrict__ Wg,
                                                    float* __restrict__ Pg) {
  __shared__ float Aug[64 * 128];
  __shared__ float Msh[64 * 64];
  __shared__ float fac[64];
  const int tid = threadIdx.x;

  for (int pass = 0; pass < 2; ++pass) {
    for (int idx = tid; idx < 64 * 64; idx += 256) {
      int r = idx >> 6, c = idx & 63;
      float a = (pass == 0) ? Minv[idx] : (Msh[idx] + ((r == c) ? KREG : 0.0f));
      Aug[r * 128 + c] = a;
      Aug[r * 128 + 64 + c] = (r == c) ? 1.0f : 0.0f;
    }
    __syncthreads();
    for (int p = 0; p < 64; ++p) {
      float pivinv = 1.0f / Aug[p * 128 + p];
      __syncthreads();
      if (tid < 128) Aug[p * 128 + tid] *= pivinv;
      __syncthreads();
      if (tid < 64) fac[tid] = Aug[tid * 128 + p];
      __syncthreads();
      for (int idx = tid; idx < 64 * 128; idx += 256) {
        int r = idx >> 7, c = idx & 127;
        if (r != p) Aug[r * 128 + c] -= fac[r] * Aug[p * 128 + c];
      }
      __syncthreads();
    }
    if (pass == 0) {
      for (int idx = tid; idx < 64 * 64; idx += 256) {
        int r = idx >> 6, c = idx & 63;
        Msh[idx] = Aug[r * 128 + 64 + c];
      }
      __syncthreads();
    }
  }
  // Aug right half now holds W.  P = W * M ; write W and P to workspace.
  for (int idx = tid; idx < 64 * 64; idx += 256) {
    int r = idx >> 6, c = idx & 63;
    float acc = 0.0f;
    for (int a = 0; a < 64; ++a) acc += Aug[r * 128 + 64 + a] * Msh[a * 64 + c];
    Pg[idx] = acc;
    Wg[idx] = Aug[r * 128 + 64 + c];
  }
}

// ---------------------------------------------------------------------------
// Kernel 2: one workgroup (8 waves) per batch element.
// W is staged into LDS with one TENSOR_LOAD_TO_LDS DMA (TDM).  The 128x128
// Schur matrix is built in LDS and factored with a blocked LU whose trailing
// updates run on V_WMMA_F32_16X16X4_F32.  Column panels are stored as -L so
// the WMMA chain needs no negation VALU ops.
// ---------------------------------------------------------------------------
__global__ __launch_bounds__(256) void solve_kernel(
    const float* __restrict__ x, const float* __restrict__ v,
    const int* __restrict__ cld, const float* __restrict__ dist,
    const float* __restrict__ mu, const float* __restrict__ cor,
    const float* __restrict__ Wg, const float* __restrict__ Pg,
    float* __restrict__ out) {
  __shared__ float Ssh[128 * SS];
  __shared__ float Wsh[64 * 64];
  __shared__ float enx[32], eny[32];
  __shared__ int cis[32], cjs[32];
  __shared__ float rhs[128];
  __shared__ float vv[128];   // v reshaped [64][2]
  __shared__ float yv[128];   // P @ v
  __shared__ float t2[128];   // G^T * lambda reshaped [64][2]
  __shared__ float fxs[32], fys[32];

  const int tid = threadIdx.x;
  const int b = blockIdx.x;

  // ---- stage W into LDS via the Tensor Data Mover (wave 0 issues) ---------
  if (tid < 32) {
    unsigned long long ga = (unsigned long long)(uintptr_t)Wg;
    unsigned int lds_off = (unsigned int)(uintptr_t)(&Wsh[0]);  // low 32 = LDS byte addr
    u32x4 g0;
    g0[0] = 1u;                                   // count=1, user descriptor
    g0[1] = lds_off;                              // lds_addr
    g0[2] = (unsigned int)ga;                     // global_addr[31:0]
    g0[3] = (unsigned int)((ga >> 32) & 0x01FFFFFFu) | (2u << 30);  // addr[56:32], type=2
    i32x8 g1;
    g1[0] = 0x00020000;   // workgroup_mask=0, data_size=2 (4B)
    g1[1] = 0x10000000;   // tensor_dim0[15:0]=4096 at bits 63:48
    g1[2] = 0x00010000;   // tensor_dim1[15:0]=1 at bits 111:80
    g1[3] = 0x10000000;   // tile_dim0=4096 at bits 127:112
    g1[4] = 0x00000001;   // tile_dim1=1
    g1[5] = 0x00001000;   // tensor_dim0_stride=4096 (low 32)
    g1[6] = 0x10000000;   // tensor_dim1_stride[15:0]=4096 at bits 223:208
    g1[7] = 0;
    i32x4 gz4 = {0, 0, 0, 0};
    i32x8 gz8 = {0, 0, 0, 0, 0, 0, 0, 0};
    __builtin_amdgcn_tensor_load_to_lds(g0, g1, gz4, gz4, gz8, 0);
    __builtin_amdgcn_s_wait_tensorcnt((short)0);
  }

  // ---- stage the small per-batch inputs -----------------------------------
  if (tid < 128) vv[tid] = v[b * 128 + tid];
  if (tid < 32) {
    int i = cld[b * 64 + 2 * tid];
    int j = cld[b * 64 + 2 * tid + 1];
    cis[tid] = i;
    cjs[tid] = j;
    float dx = x[b * 128 + 2 * j] - x[b * 128 + 2 * i];
    float dy = x[b * 128 + 2 * j + 1] - x[b * 128 + 2 * i + 1];
    float nrm = sqrtf(dx * dx + dy * dy);
    enx[tid] = dx / nrm;
    eny[tid] = dy / nrm;
  }
  for (int idx = tid; idx < 128 * 128; idx += 256)
    Ssh[(idx >> 7) * SS + (idx & 127)] = 0.0f;
  __syncthreads();

  // ---- build S = reg*I - F - G A^{-1} G^T  (closed form via W) ------------
  for (int idx = tid; idx < 1024; idx += 256) {
    int r = idx >> 5, c = idx & 31;
    int Ir = cis[r], Jr = cjs[r], Ic = cis[c], Jc = cjs[c];
    float g = Wsh[Ir * 64 + Ic] - Wsh[Ir * 64 + Jc] - Wsh[Jr * 64 + Ic] + Wsh[Jr * 64 + Jc];
    float D = enx[r] * enx[c] + eny[r] * eny[c];   // e_n . e_n' == e_t . e_t'
    float X = eny[r] * enx[c] - enx[r] * eny[c];   // e_n . e_t'
    float Dg = D * g, Xg = X * g;
    Ssh[r * SS + c] = -Dg;
    Ssh[r * SS + 32 + 2 * c] = -Xg;
    Ssh[r * SS + 33 + 2 * c] = Xg;
    Ssh[(32 + 2 * r) * SS + c] = Xg;
    Ssh[(33 + 2 * r) * SS + c] = -Xg;
    Ssh[(32 + 2 * r) * SS + 32 + 2 * c] = -Dg;
    Ssh[(32 + 2 * r) * SS + 33 + 2 * c] = Dg;
    Ssh[(33 + 2 * r) * SS + 32 + 2 * c] = Dg;
    Ssh[(33 + 2 * r) * SS + 33 + 2 * c] = -Dg;
  }
  if (tid < 32) {   // -F entries (E blocks and mu diag)
    int k = tid;
    Ssh[(96 + k) * SS + k] = -mu[b * 32 + k];
    Ssh[(32 + 2 * k) * SS + 96 + k] = -1.0f;
    Ssh[(33 + 2 * k) * SS + 96 + k] = -1.0f;
    Ssh[(96 + k) * SS + 32 + 2 * k] = 1.0f;
    Ssh[(96 + k) * SS + 33 + 2 * k] = 1.0f;
  }
  __syncthreads();
  if (tid < 128) Ssh[tid * SS + tid] += KREG;

  // ---- rhs2 = -h - G*y with y = -kron(P,I2)*v -----------------------------
  if (tid < 128) {   // yv = P @ vv  (kron structure)
    int m = tid >> 1, cpt = tid & 1;
    float acc = 0.0f;
    for (int a = 0; a < 64; ++a) acc += Pg[m * 64 + a] * vv[2 * a + cpt];
    yv[tid] = acc;
  }
  __syncthreads();
  if (tid < 32) {
    int c = tid, i = cis[c], j = cjs[c];
    float ex = enx[c], ey = eny[c];
    float tx = -ey, ty = ex;
    float rvx = vv[2 * j] - vv[2 * i], rvy = vv[2 * j + 1] - vv[2 * i + 1];
    float Jcnv = ex * rvx + ey * rvy;
    float hn = fminf(dist[b * 32 + c] * 12.5f, cor[b * 32 + c] * Jcnv);
    float dyx = yv[2 * j] - yv[2 * i], dyy = yv[2 * j + 1] - yv[2 * i + 1];
    float gn = ex * dyx + ey * dyy;   // e_n . (yv_j - yv_i);  y = -yv
    float gt = tx * dyx + ty * dyy;
    rhs[c] = -hn + gn;
    rhs[32 + 2 * c] = gt;
    rhs[33 + 2 * c] = -gt;
    rhs[96 + c] = 0.0f;
  }
  __syncthreads();

  // ---- blocked LU (no pivoting); column panels hold -L --------------------
  for (int k = 0; k < 8; ++k) {
    const int kb = 16 * k;
    // (a) factor 16x16 diagonal block (multipliers stored positive)
    for (int p = 0; p < 16; ++p) {
      __syncthreads();
      if (tid > p && tid < 16)
        Ssh[(kb + tid) * SS + kb + p] /= Ssh[(kb + p) * SS + kb + p];
      __syncthreads();
      int r = tid >> 4, c = tid & 15;
      if (r > p && c > p)
        Ssh[(kb + r) * SS + kb + c] -=
            Ssh[(kb + r) * SS + kb + p] * Ssh[(kb + p) * SS + kb + c];
    }
    if (k == 7) { __syncthreads(); break; }
    const int w = 112 - 16 * k;
    // (b) row panel: U_kj = L_kk^{-1} A_kj
    for (int p = 0; p < 15; ++p) {
      __syncthreads();
      int total = (15 - p) * w;
      for (int idx = tid; idx < total; idx += 256) {
        int r = p + 1 + idx / w;
        int c = kb + 16 + idx % w;
        Ssh[(kb + r) * SS + c] -= Ssh[(kb + r) * SS + kb + p] * Ssh[(kb + p) * SS + c];
      }
    }
    // (c) column panel stored NEGATED: s_ik = -A_ik U_kk^{-1}
    for (int p = 0; p < 16; ++p) {
      __syncthreads();
      float nud = -Ssh[(kb + p) * SS + kb + p];
      for (int r = kb + 16 + tid; r < 128; r += 256)
        Ssh[r * SS + kb + p] /= nud;
      __syncthreads();
      int ncols = 15 - p;
      int total = w * ncols;
      for (int idx = tid; idx < total; idx += 256) {
        int r = kb + 16 + idx / ncols;
        int c = p + 1 + idx % ncols;
        Ssh[r * SS + kb + c] += Ssh[r * SS + kb + p] * Ssh[(kb + p) * SS + kb + c];
      }
    }
    __syncthreads();
    // (d) trailing update A_ij += (-L_ik) * U_kj via V_WMMA_F32_16X16X4_F32
    const int nt = 7 - k;
    const int wave = tid >> 5;
    const int lane = tid & 31;
    const int m16 = lane & 15;
    const int h = lane >> 4;
    for (int t = wave; t < nt * nt; t += 8) {
      const int gi = kb + 16 + 16 * (t / nt);
      const int gj = kb + 16 + 16 * (t % nt);
      v8f cfrag;
#pragma unroll
      for (int r = 0; r < 8; ++r)
        cfrag[r] = Ssh[(gi + r + 8 * h) * SS + gj + m16];
#pragma unroll
      for (int q = 0; q < 4; ++q) {
        v2f a = *(const v2f*)&Ssh[(gi + m16) * SS + kb + 4 * q + 2 * h];  // -L, contiguous
        v2f bfrag;
        bfrag[0] = Ssh[(kb + 4 * q + 2 * h) * SS + gj + m16];
        bfrag[1] = Ssh[(kb + 4 * q + 2 * h + 1) * SS + gj + m16];
        cfrag = __builtin_amdgcn_wmma_f32_16x16x4_f32(
            false, a, false, bfrag, (short)0, cfrag, false, false);
      }
#pragma unroll
      for (int r = 0; r < 8; ++r)
        Ssh[(gi + r + 8 * h) * SS + gj + m16] = cfrag[r];
    }
    __syncthreads();
  }

  // ---- triangular solves on the single rhs --------------------------------
  for (int col = 0; col < 127; ++col) {   // L (unit lower; below-block = -L)
    __syncthreads();
    if (tid > col && tid < 128) {
      float u = Ssh[tid * SS + col] * rhs[col];
      rhs[tid] -= ((tid >> 4) == (col >> 4)) ? u : -u;
    }
  }
  for (int col = 127; col >= 0; --col) {  // U
    __syncthreads();
    if (tid == col) rhs[col] /= Ssh[col * SS + col];
    __syncthreads();
    if (tid < col) rhs[tid] -= Ssh[tid * SS + col] * rhs[col];
  }
  __syncthreads();

  // ---- t2 = G^T lambda (deterministic scan), v+ = P v + W t2 --------------
  if (tid < 32) {
    int c = tid;
    float ln = rhs[c];
    float lt = rhs[32 + 2 * c] - rhs[33 + 2 * c];
    float ex = enx[c], ey = eny[c];
    fxs[c] = ex * ln + (-ey) * lt;
    fys[c] = ey * ln + ex * lt;
  }
  __syncthreads();
  if (tid < 128) {
    int a = tid >> 1, cpt = tid & 1;
    float acc = 0.0f;
    for (int c = 0; c < 32; ++c) {
      float f = cpt ? fys[c] : fxs[c];
      if (cjs[c] == a) acc += f;
      if (cis[c] == a) acc -= f;
    }
    t2[tid] = acc;
  }
  __syncthreads();
  if (tid < 128) {
    int m = tid >> 1, cpt = tid & 1;
    float acc = yv[tid];   // P @ v part
    for (int a = 0; a < 64; ++a) acc += Wsh[m * 64 + a] * t2[2 * a + cpt];
    out[b * 256 + 128 + tid] = acc;
    out[b * 256 + tid] = x[b * 128 + tid];
  }
}

extern "C" void kernel_launch(void* const* d_in, const int* in_sizes, int n_in,
                              void* d_out, int out_size, void* d_ws, size_t ws_size,
                              hipStream_t stream) {
  (void)in_sizes; (void)n_in; (void)out_size; (void)ws_size;
  const float* x    = (const float*)d_in[0];
  const float* v    = (const float*)d_in[1];
  const int*   cld  = (const int*)d_in[2];
  const float* dist = (const float*)d_in[3];
  const float* mu   = (const float*)d_in[4];
  const float* cor  = (const float*)d_in[5];
  const float* Minv = (const float*)d_in[6];
  float* out = (float*)d_out;
  float* Wg = (float*)d_ws;          // 64*64 floats
  float* Pg = Wg + 64 * 64;          // 64*64 floats

  setup_kernel<<<1, 256, 0, stream>>>(Minv, Wg, Pg);
  solve_kernel<<<1024, 256, 0, stream>>>(x, v, cld, dist, mu, cor, Wg, Pg, out);
}